// H36MHuman_8091718385991
// MI455X (gfx1250) — compile-verified
//
#include <hip/hip_runtime.h>
#include <hip/hip_bf16.h>
#include <stdint.h>
#include <stddef.h>

#ifndef __has_builtin
#define __has_builtin(x) 0
#endif

#if __has_builtin(__builtin_amdgcn_global_load_async_to_lds_b128)
#define FK_ASYNC 1
#else
#define FK_ASYNC 0
#endif
#if __has_builtin(__builtin_amdgcn_s_wait_asynccnt)
#define FK_WAITB 1
#else
#define FK_WAITB 0
#endif

#define FPB 128                   // frames (threads) per block
#define N_STAGE 8                 // 8 stages x 4 joints = 32 joints
#define FRAME_BYTES 1152          // 32 joints * 9 f32
#define STAGE_FRAME_BYTES 144     // 4 joints * 9 f32
#define CHUNKS_PER_THREAD 9       // (FPB*144/16)/FPB
#define LDS_FRAME_DW 44           // 36 data dwords + 8 pad -> 176 B (16B mult, 2-way banks max)

typedef int v4i __attribute__((vector_size(16)));   // matches builtin param type

__device__ __forceinline__ void copy_chunk16(const void* g, void* l) {
#if FK_ASYNC
  // CDNA5 async data mover: 16B global -> LDS, tracked by ASYNCcnt, no VGPR round-trip
  __builtin_amdgcn_global_load_async_to_lds_b128((v4i*)g, (v4i*)l, 0, 0);
#else
  *(float4*)l = *(const float4*)g;
#endif
}

template <int N>
__device__ __forceinline__ void wait_async() {
#if FK_ASYNC
#if FK_WAITB
  __builtin_amdgcn_s_wait_asynccnt(N);
#else
  asm volatile("s_wait_asynccnt %0" ::"i"(N) : "memory");
#endif
#endif
}

// Stage s copies FPB frames x 144B (joints 4s..4s+3) into ldsbuf, coalesced in 16B chunks.
__device__ __forceinline__ void issue_stage(const char* rotc, float* ldsbuf, int s,
                                            int tid, int frame0, int n) {
  char* lbase = (char*)ldsbuf;
#pragma unroll
  for (int k = 0; k < CHUNKS_PER_THREAD; ++k) {
    int c = tid + k * FPB;          // linear chunk id in [0, FPB*9)
    int f = c / 9;                  // frame within block
    int w = c - f * 9;              // 16B chunk within the 144B stage slice
    int fg = frame0 + f;
    if (fg > n - 1) fg = n - 1;     // clamp OOB frames (redundant reads, never OOB)
    const char* g = rotc + (size_t)fg * FRAME_BYTES + s * STAGE_FRAME_BYTES + w * 16;
    char* l = lbase + f * (LDS_FRAME_DW * 4) + w * 16;
    copy_chunk16(g, l);
  }
}

__device__ __forceinline__ void loadR(const float* fb, int jj, float R[9]) {
#pragma unroll
  for (int k = 0; k < 9; ++k) R[k] = fb[jj * 9 + k];
}

__device__ __forceinline__ void storeP(float* outp, int j, const float p[3], bool st) {
  if (st) { outp[3*j+0] = p[0]; outp[3*j+1] = p[1]; outp[3*j+2] = p[2]; }
}

// joint whose parent is joint 0: treated as chain root (reference `p > 0` branch)
__device__ __forceinline__ void j_root(int jj, int j, const float* fb,
                                       const float* __restrict__ off,
                                       float Rc[9], float pc[3], float* outp, bool st) {
  loadR(fb, jj, Rc);
  pc[0] = off[3*j]; pc[1] = off[3*j+1]; pc[2] = off[3*j+2];
  storeP(outp, j, pc, st);
}

// interior joint: p = off_j @ Rp + pp ; Rd = R_j @ Rp   (row-vector convention)
__device__ __forceinline__ void j_mul(int jj, int j, const float* fb,
                                      const float* __restrict__ off,
                                      const float Rp[9], const float pp[3],
                                      float Rd[9], float pd[3], float* outp, bool st) {
  float Rl[9];
  loadR(fb, jj, Rl);
  const float o0 = off[3*j], o1 = off[3*j+1], o2 = off[3*j+2];
  float pn[3];
#pragma unroll
  for (int k = 0; k < 3; ++k)
    pn[k] = fmaf(o0, Rp[k], fmaf(o1, Rp[3+k], fmaf(o2, Rp[6+k], pp[k])));
  float Rn[9];
#pragma unroll
  for (int i = 0; i < 3; ++i)
#pragma unroll
    for (int k = 0; k < 3; ++k)
      Rn[3*i+k] = fmaf(Rl[3*i+0], Rp[k], fmaf(Rl[3*i+1], Rp[3+k], Rl[3*i+2] * Rp[6+k]));
#pragma unroll
  for (int k = 0; k < 9; ++k) Rd[k] = Rn[k];
  pd[0] = pn[0]; pd[1] = pn[1]; pd[2] = pn[2];
  storeP(outp, j, pn, st);
}

// leaf joint: position only, no rotation accumulate (its R is never consumed)
__device__ __forceinline__ void j_leaf(int j, const float* __restrict__ off,
                                       const float Rp[9], const float pp[3],
                                       float* outp, bool st) {
  const float o0 = off[3*j], o1 = off[3*j+1], o2 = off[3*j+2];
  float pn[3];
#pragma unroll
  for (int k = 0; k < 3; ++k)
    pn[k] = fmaf(o0, Rp[k], fmaf(o1, Rp[3+k], fmaf(o2, Rp[6+k], pp[k])));
  storeP(outp, j, pn, st);
}

__global__ __launch_bounds__(FPB) void h36m_fk_kernel(
    const float* __restrict__ rot, const float* __restrict__ off,
    float* __restrict__ out, int n) {
  __shared__ __align__(16) float lds[2][FPB * LDS_FRAME_DW];  // 45,056 B double buffer

  const int tid = threadIdx.x;
  const int frame0 = blockIdx.x * FPB;
  const int frame = frame0 + tid;
  const bool st = frame < n;
  const char* rotc = (const char*)rot;
  float* outp = out + (size_t)frame * 96;

  float Rc[9], pc[3];     // current chain state
  float R12[9], p12[3];   // branch joint 12 (consumed at joints 16 and 24)
  float Rb[9],  pb[3];    // branch joints 19 / 27 (disjoint lifetimes)

  // prologue: prime both LDS buffers
  issue_stage(rotc, lds[0], 0, tid, frame0, n);
  issue_stage(rotc, lds[1], 1, tid, frame0, n);

  const float* fb;
#define STAGE_BEGIN(sidx, waitn)                      \
  wait_async<waitn>();                                \
  __syncthreads();                                    \
  fb = &lds[(sidx) & 1][tid * LDS_FRAME_DW];
#define STAGE_END(sidx)                               \
  __syncthreads();                                    \
  issue_stage(rotc, lds[(sidx) & 1], (sidx) + 2, tid, frame0, n);

  // ---- stage 0: joints 0..3 ----
  STAGE_BEGIN(0, CHUNKS_PER_THREAD)
  if (st) { outp[0] = off[0]; outp[1] = off[1]; outp[2] = off[2]; }   // joint 0: fixed
  j_root(1, 1, fb, off, Rc, pc, outp, st);
  j_mul (2, 2, fb, off, Rc, pc, Rc, pc, outp, st);
  j_mul (3, 3, fb, off, Rc, pc, Rc, pc, outp, st);
  STAGE_END(0)

  // ---- stage 1: joints 4..7 ----
  STAGE_BEGIN(1, CHUNKS_PER_THREAD)
  j_mul (0, 4, fb, off, Rc, pc, Rc, pc, outp, st);
  j_leaf(5, off, Rc, pc, outp, st);
  j_root(2, 6, fb, off, Rc, pc, outp, st);
  j_mul (3, 7, fb, off, Rc, pc, Rc, pc, outp, st);
  STAGE_END(1)

  // ---- stage 2: joints 8..11 ----
  STAGE_BEGIN(2, CHUNKS_PER_THREAD)
  j_mul (0, 8,  fb, off, Rc, pc, Rc, pc, outp, st);
  j_mul (1, 9,  fb, off, Rc, pc, Rc, pc, outp, st);
  j_leaf(10, off, Rc, pc, outp, st);
  j_root(3, 11, fb, off, Rc, pc, outp, st);
  STAGE_END(2)

  // ---- stage 3: joints 12..15 ----
  STAGE_BEGIN(3, CHUNKS_PER_THREAD)
  j_mul (0, 12, fb, off, Rc, pc, Rc, pc, outp, st);
#pragma unroll
  for (int k = 0; k < 9; ++k) R12[k] = Rc[k];
  p12[0] = pc[0]; p12[1] = pc[1]; p12[2] = pc[2];
  j_mul (1, 13, fb, off, Rc, pc, Rc, pc, outp, st);
  j_mul (2, 14, fb, off, Rc, pc, Rc, pc, outp, st);
  j_leaf(15, off, Rc, pc, outp, st);
  STAGE_END(3)

  // ---- stage 4: joints 16..19 ----
  STAGE_BEGIN(4, CHUNKS_PER_THREAD)
  j_mul (0, 16, fb, off, R12, p12, Rc, pc, outp, st);
  j_mul (1, 17, fb, off, Rc, pc, Rc, pc, outp, st);
  j_mul (2, 18, fb, off, Rc, pc, Rc, pc, outp, st);
  j_mul (3, 19, fb, off, Rc, pc, Rc, pc, outp, st);
#pragma unroll
  for (int k = 0; k < 9; ++k) Rb[k] = Rc[k];
  pb[0] = pc[0]; pb[1] = pc[1]; pb[2] = pc[2];
  STAGE_END(4)

  // ---- stage 5: joints 20..23 ----
  STAGE_BEGIN(5, CHUNKS_PER_THREAD)
  j_mul (0, 20, fb, off, Rc, pc, Rc, pc, outp, st);
  j_leaf(21, off, Rc, pc, outp, st);
  j_mul (2, 22, fb, off, Rb, pb, Rc, pc, outp, st);
  j_leaf(23, off, Rc, pc, outp, st);
  STAGE_END(5)

  // ---- stage 6: joints 24..27 ----
  STAGE_BEGIN(6, CHUNKS_PER_THREAD)
  j_mul (0, 24, fb, off, R12, p12, Rc, pc, outp, st);
  j_mul (1, 25, fb, off, Rc, pc, Rc, pc, outp, st);
  j_mul (2, 26, fb, off, Rc, pc, Rc, pc, outp, st);
  j_mul (3, 27, fb, off, Rc, pc, Rc, pc, outp, st);
#pragma unroll
  for (int k = 0; k < 9; ++k) Rb[k] = Rc[k];
  pb[0] = pc[0]; pb[1] = pc[1]; pb[2] = pc[2];
  // no further issues: last stage already in flight

  // ---- stage 7: joints 28..31 ----
  wait_async<0>();
  __syncthreads();
  fb = &lds[7 & 1][tid * LDS_FRAME_DW];
  j_mul (0, 28, fb, off, Rc, pc, Rc, pc, outp, st);
  j_leaf(29, off, Rc, pc, outp, st);
  j_mul (2, 30, fb, off, Rb, pb, Rc, pc, outp, st);
  j_leaf(31, off, Rc, pc, outp, st);
#undef STAGE_BEGIN
#undef STAGE_END
}

extern "C" void kernel_launch(void* const* d_in, const int* in_sizes, int n_in,
                              void* d_out, int out_size, void* d_ws, size_t ws_size,
                              hipStream_t stream) {
  const float* rot = (const float*)d_in[0];   // (n, 32, 3, 3) f32
  const float* off = (const float*)d_in[1];   // (32, 3) f32
  float* out = (float*)d_out;                 // (n, 32, 3) f32
  const int n = in_sizes[0] / 288;
  const int blocks = (n + FPB - 1) / FPB;
  h36m_fk_kernel<<<blocks, FPB, 0, stream>>>(rot, off, out, n);
}